// MoEGate_71829033058634
// MI455X (gfx1250) — compile-verified
//
#include <hip/hip_runtime.h>
#include <hip/hip_bf16.h>
#include <math.h>

typedef __attribute__((ext_vector_type(2))) float v2f;
typedef __attribute__((ext_vector_type(8))) float v8f;

#define HDIM   4096
#define NEXP   64
#define TOPK   8
#define NBATCH 4
#define SEQ    4096
#define NTOK   (NBATCH * SEQ)
#define KC     64            // K-chunk of weight staged in LDS
#define WPAD   4
#define WSTRIDE (KC + WPAD)  // 68 dwords -> conflict-free ds_load_b64

// ---------------------------------------------------------------------------
// Kernel 1: logits[t][e] = sum_h x[t][h] * w[e][h]   via V_WMMA_F32_16X16X4_F32
// block = 256 threads (8 waves); wave -> 16 tokens x 64 experts
// ---------------------------------------------------------------------------
__global__ __launch_bounds__(256) void gate_gemm(const float* __restrict__ x,
                                                 const float* __restrict__ w,
                                                 float* __restrict__ logits) {
    __shared__ __align__(16) float wLds[NEXP * WSTRIDE];

    const int tid  = threadIdx.x;
    const int wave = tid >> 5;
    const int lane = tid & 31;
    const int m    = lane & 15;   // token row within tile / expert col within tile
    const int h    = lane >> 4;   // half-wave selector (K = v + 2*h)
    const int tok0 = blockIdx.x * 128 + wave * 16;

    const float* aRow = x + (size_t)(tok0 + m) * HDIM + 2 * h;

    v8f acc[4];
#pragma unroll
    for (int n = 0; n < 4; ++n) acc[n] = (v8f){0.f,0.f,0.f,0.f,0.f,0.f,0.f,0.f};

    for (int kc = 0; kc < HDIM; kc += KC) {
        if (kc) __syncthreads();              // previous chunk fully consumed
        // coalesced stage of weight chunk: 64 experts x KC floats
#pragma unroll
        for (int i = tid; i < NEXP * (KC / 4); i += 256) {
            const int e  = i >> 4;            // KC/4 == 16
            const int kq = i & 15;
            const float4 v = *(const float4*)(w + (size_t)e * HDIM + kc + kq * 4);
            *(float4*)&wLds[e * WSTRIDE + kq * 4] = v;
        }
        __syncthreads();

        // prefetch next weight chunk into caches (global_prefetch_b8)
        if (kc + KC < HDIM) {
            const int e = tid >> 2, q = tid & 3;     // 64 rows x 4 cachelines
            __builtin_prefetch(w + (size_t)e * HDIM + kc + KC + q * 16, 0, 1);
        }

#pragma unroll
        for (int ks = 0; ks < KC; ks += 4) {
            // A fragment: 16x4 fp32, lane(m,h) holds K = ks+2h, ks+2h+1
            v2f a = *(const v2f*)(aRow + kc + ks);
#pragma unroll
            for (int n = 0; n < 4; ++n) {
                const int e = n * 16 + m;     // B col = expert
                v2f b = *(const v2f*)&wLds[e * WSTRIDE + ks + 2 * h];
                acc[n] = __builtin_amdgcn_wmma_f32_16x16x4_f32(
                    /*neg_a=*/false, a, /*neg_b=*/false, b,
                    /*c_mod=*/(short)0, acc[n],
                    /*reuse_a=*/false, /*reuse_b=*/false);
            }
        }
    }

    // C/D layout: vgpr r, half h -> M = r + 8h ; lane%16 -> N
#pragma unroll
    for (int n = 0; n < 4; ++n)
#pragma unroll
        for (int r = 0; r < 8; ++r)
            logits[(size_t)(tok0 + r + 8 * h) * NEXP + n * 16 + m] = acc[n][r];
}

// ---------------------------------------------------------------------------
// Kernel 2: per-token softmax, biased top-8, renorm; LDS-accumulated ce /
// mean-score partials flushed with global float atomics.
// ---------------------------------------------------------------------------
__global__ __launch_bounds__(256) void gate_topk(const float* __restrict__ logits,
                                                 const float* __restrict__ bias,
                                                 int* __restrict__ idxOut,
                                                 float* __restrict__ wOut,
                                                 float* __restrict__ gMean,
                                                 float* __restrict__ gCe) {
    __shared__ float sBias[NEXP];
    __shared__ float sMean[NEXP];
    __shared__ float sCe[NEXP];

    const int tid = threadIdx.x;
    if (tid < NEXP) { sBias[tid] = bias[tid]; sMean[tid] = 0.f; sCe[tid] = 0.f; }
    __syncthreads();

    const int t = blockIdx.x * 256 + tid;
    const int b = t >> 12;                    // SEQ = 4096 tokens per batch

    float l[NEXP];
    const float4* lp = (const float4*)(logits + (size_t)t * NEXP);
#pragma unroll
    for (int i = 0; i < NEXP / 4; ++i) {
        float4 v = lp[i];
        l[4 * i + 0] = v.x; l[4 * i + 1] = v.y;
        l[4 * i + 2] = v.z; l[4 * i + 3] = v.w;
    }

    float mx = -INFINITY;
#pragma unroll
    for (int e = 0; e < NEXP; ++e) mx = fmaxf(mx, l[e]);
    float sum = 0.f;
#pragma unroll
    for (int e = 0; e < NEXP; ++e) { float ex = __expf(l[e] - mx); l[e] = ex; sum += ex; }
    const float inv = 1.0f / sum;
#pragma unroll
    for (int e = 0; e < NEXP; ++e) {
        l[e] *= inv;                          // original softmax score
        atomicAdd(&sMean[e], l[e]);           // ds_add_f32
        l[e] += sBias[e];                     // biased routing score
    }

    // top-8 over biased scores; recover original score as biased - bias
    unsigned long long mask = 0ull;
    int   isel[TOPK];
    float wsel[TOPK];
    float wsum = 0.f;
#pragma unroll
    for (int k = 0; k < TOPK; ++k) {
        float best = -INFINITY; int bi = 0;
#pragma unroll
        for (int e = 0; e < NEXP; ++e) {
            bool free = !((mask >> e) & 1ull);
            if (free && l[e] > best) { best = l[e]; bi = e; }
        }
        mask |= (1ull << bi);
        const float sc = best - sBias[bi];
        isel[k] = bi; wsel[k] = sc; wsum += sc;
        atomicAdd(&sCe[bi], 1.0f);
    }

    const float rinv = 1.0f / (wsum + 1e-20f);
#pragma unroll
    for (int k = 0; k < TOPK; ++k) {
        idxOut[(size_t)t * TOPK + k] = isel[k];
        wOut [(size_t)t * TOPK + k] = wsel[k] * rinv;
    }

    __syncthreads();
    if (tid < NEXP) {
        atomicAdd(&gMean[b * NEXP + tid], sMean[tid]);
        atomicAdd(&gCe  [b * NEXP + tid], sCe[tid]);
    }
}

// ---------------------------------------------------------------------------
// Kernel 3: aux loss scalar
// ---------------------------------------------------------------------------
__global__ void gate_aux(const float* __restrict__ gMean,
                         const float* __restrict__ gCe,
                         float* __restrict__ auxOut) {
    __shared__ float red[NEXP];
    const int e = threadIdx.x;                // 64 threads
    float v = 0.f;
#pragma unroll
    for (int b = 0; b < NBATCH; ++b) v += gCe[b * NEXP + e] * gMean[b * NEXP + e];
    red[e] = v;
    __syncthreads();
    for (int s = 32; s > 0; s >>= 1) {
        if (e < s) red[e] += red[e + s];
        __syncthreads();
    }
    if (e == 0) {
        // ALPHA * (1/B) * sum_b sum_e [ce * E/(S*K)] * [score_sum / S]
        const float factor = 1e-3f * ((float)NEXP / ((float)SEQ * TOPK))
                             / (float)SEQ / (float)NBATCH;
        auxOut[0] = red[0] * factor;
    }
}

__global__ void zero_acc(float* __restrict__ p, int n) {
    int i = blockIdx.x * blockDim.x + threadIdx.x;
    if (i < n) p[i] = 0.f;
}

// ---------------------------------------------------------------------------
extern "C" void kernel_launch(void* const* d_in, const int* in_sizes, int n_in,
                              void* d_out, int out_size, void* d_ws, size_t ws_size,
                              hipStream_t stream) {
    const float* x    = (const float*)d_in[0];   // [4,4096,4096] fp32
    const float* w    = (const float*)d_in[1];   // [64,4096]     fp32
    const float* bias = (const float*)d_in[2];   // [64]          fp32

    float* logits = (float*)d_ws;                           // 16384*64 fp32 = 4 MB
    float* gMean  = logits + (size_t)NTOK * NEXP;           // [4][64]
    float* gCe    = gMean + NBATCH * NEXP;                  // [4][64]

    int*   idxOut = (int*)d_out;                            // top-8 indices (int32 bits)
    float* wOut   = (float*)d_out + (size_t)NTOK * TOPK;    // top-8 weights
    float* auxOut = (float*)d_out + (size_t)NTOK * TOPK * 2;// aux loss scalar

    zero_acc <<<1, 2 * NBATCH * NEXP, 0, stream>>>(gMean, 2 * NBATCH * NEXP);
    gate_gemm<<<NTOK / 128, 256, 0, stream>>>(x, w, logits);
    gate_topk<<<NTOK / 256, 256, 0, stream>>>(logits, bias, idxOut, wOut, gMean, gCe);
    gate_aux <<<1, NEXP, 0, stream>>>(gMean, gCe, auxOut);
}